// PSAMask_23845658427871
// MI455X (gfx1250) — compile-verified
//
#include <hip/hip_runtime.h>
#include <stdint.h>

// PSAMask "collect", H=W=h_mask=w_mask=97, B=1.
// out[nh*97+nw, h, w] = x[(nh-h+48)*97+(nw-w+48), h, w], zero outside mask.
// Pure data movement: ~354MB in + ~354MB out -> ~30us at 23.3 TB/s.
// One block per (nh,h); input tile x[dh*97+dw, h, :] (97x97 floats, 37.6KB)
// staged into LDS by the Tensor Data Mover (iterate mode gives LDS pitch 98
// floats -> conflict-free shear reads), then coalesced row stores.

#define HM      97
#define HW      9409      // 97*97
#define HALF    48
#define PITCH   98        // LDS row pitch in floats (odd delta mod 64 banks)
#define NTHREADS 256

#if defined(__has_builtin)
#  if __has_builtin(__builtin_amdgcn_tensor_load_to_lds) && __has_builtin(__builtin_amdgcn_s_wait_tensorcnt)
#    define USE_TDM 1
#  endif
#endif

typedef unsigned int u32x4 __attribute__((ext_vector_type(4)));
typedef int          i32x4 __attribute__((ext_vector_type(4)));
typedef int          i32x8 __attribute__((ext_vector_type(8)));

__global__ __launch_bounds__(NTHREADS)
void PSAMask_23845658427871_kernel(const float* __restrict__ x,
                                   float* __restrict__ out) {
    __shared__ float tile[HM * PITCH];   // 97 rows, pitch 98 floats = 38KB

    const int nh  = blockIdx.x;
    const int h   = blockIdx.y;
    const int b   = blockIdx.z;
    const int tid = threadIdx.x;

    const int dh = nh - h + HALF;                     // block-uniform
    const bool valid = ((unsigned)dh < (unsigned)HM);

    const size_t batch_off = (size_t)b * (size_t)HW * (size_t)HW;
    // out[b, nh*97 + nw, h, w] -> out_base + nw*9409 + w
    const size_t out_base = batch_off + ((size_t)nh * HM) * (size_t)HW + (size_t)h * HM;

    if (valid) {
        // tile[dw][w] = x[b, dh*97+dw, h, w]; rows 388B, row stride 9409 floats
        const float* src = x + batch_off + ((size_t)dh * HM) * (size_t)HW + (size_t)h * HM;

#ifdef USE_TDM
        if (tid < 32) {   // one wave drives the Tensor Data Mover
            const uint64_t gaddr = (uint64_t)(uintptr_t)src;
            const unsigned lds0  = (unsigned)(uintptr_t)(&tile[0]); // low 32b = LDS offset

            // D# group 0: count=1 valid, lds_addr, 57-bit global_addr, type=2
            u32x4 g0;
            g0.x = 1u;
            g0.y = lds0;
            g0.z = (unsigned)(gaddr & 0xffffffffull);
            g0.w = (unsigned)((gaddr >> 32) & 0x01ffffffull) | (2u << 30);

            // D# group 1: data_size=4B (code 2), iterate_enable=1, wg_mask=0,
            // tensor_dim0=97, tensor_dim1=1, tile_dim0=97, tile_dim1=1,
            // tensor_dim0_stride=9409 elements.
            i32x8 g1;
            g1[0] = (int)((2u << 16) | (1u << 19));
            g1[1] = (int)((unsigned)HM << 16);   // tensor_dim0 lo16 at bits[63:48]
            g1[2] = (int)(1u << 16);             // tensor_dim1 lo16 at bits[95:80]
            g1[3] = (int)((unsigned)HM << 16);   // tile_dim0 at bits[127:112]
            g1[4] = 1;                           // tile_dim1 at bits[143:128]
            g1[5] = HW;                          // tensor_dim0_stride lo32
            g1[6] = 0;
            g1[7] = 0;

            // D# group 2 (iterate mode): lds_addr_increment = 98 floats (bytes),
            // global_addr_increment = 9409 (data units), iterate_count = 96 (=97x)
            i32x4 g2;
            g2.x = 0;
            g2.y = PITCH * 4;
            g2.z = HW;
            g2.w = (int)((unsigned)(HM - 1) << 16);

            i32x4 g3 = {0, 0, 0, 0};

            // clang-23 / therock-10.0 lane: 6-arg form
            // (u32x4 g0, i32x8 g1, i32x4 g2, i32x4 g3, i32x8 extra, i32 cpol)
            i32x8 g4 = {0, 0, 0, 0, 0, 0, 0, 0};

            __builtin_amdgcn_tensor_load_to_lds(g0, g1, g2, g3, g4, 0);
            __builtin_amdgcn_s_wait_tensorcnt(0);
        }
        __syncthreads();
#else
        for (int i = tid; i < HM * HM; i += NTHREADS) {
            int dw = i / HM;
            int w  = i - dw * HM;
            tile[dw * PITCH + w] = src[(size_t)dw * HW + w];
        }
        __syncthreads();
#endif

        // Shear-gather from LDS; consecutive lanes -> consecutive w:
        // LDS delta per lane = 1 - PITCH = -97 (odd mod 64) -> conflict-free,
        // global stores coalesced along 388B output rows.
        for (int i = tid; i < HM * HM; i += NTHREADS) {
            int nw = i / HM;
            int w  = i - nw * HM;
            int dw = nw - w + HALF;
            float v = 0.0f;
            if ((unsigned)dw < (unsigned)HM) v = tile[dw * PITCH + w];
            out[out_base + (size_t)nw * HW + w] = v;
        }
    } else {
        // whole (nh,h) slab outside the vertical mask window -> zeros
        for (int i = tid; i < HM * HM; i += NTHREADS) {
            int nw = i / HM;
            int w  = i - nw * HM;
            out[out_base + (size_t)nw * HW + w] = 0.0f;
        }
    }
}

extern "C" void kernel_launch(void* const* d_in, const int* in_sizes, int n_in,
                              void* d_out, int out_size, void* d_ws, size_t ws_size,
                              hipStream_t stream) {
    (void)n_in; (void)out_size; (void)d_ws; (void)ws_size;
    const float* x = (const float*)d_in[0];
    float* out = (float*)d_out;
    int B = in_sizes[0] / (HW * HW);
    if (B < 1) B = 1;
    dim3 grid(HM, HM, B);
    dim3 block(NTHREADS);
    hipLaunchKernelGGL(PSAMask_23845658427871_kernel, grid, block, 0, stream, x, out);
}